// ModifiedATSS_25675314495727
// MI455X (gfx1250) — compile-verified
//
#include <hip/hip_runtime.h>
#include <hip/hip_bf16.h>

typedef float v2f __attribute__((ext_vector_type(2)));
typedef float v8f __attribute__((ext_vector_type(8)));

#define B_  32
#define Q_  16384
#define G_  128
#define K_  32
#define N_  9
#define GT_TILE 16
#define QT  512          // Q strip width held in LDS per iteration
#define QTP (QT + 4)     // padded row stride (floats) to de-alias LDS banks
#define CAP 128          // per-row candidate queue capacity per strip

// Sorted ascending insert into a 32-entry (dist, idx) list held in LDS.
// Strict compares reproduce jax top_k tie order (earliest index wins)
// provided candidates are offered in increasing q order.
__device__ __forceinline__ void topk_insert(float* __restrict__ topD,
                                            int*   __restrict__ topI,
                                            float dv, int q) {
    if (dv < topD[K_ - 1]) {
        int pos = K_ - 1;
        #pragma unroll 1
        while (pos > 0 && topD[pos - 1] > dv) {
            topD[pos] = topD[pos - 1];
            topI[pos] = topI[pos - 1];
            --pos;
        }
        topD[pos] = dv;
        topI[pos] = q;
    }
}

__global__ __launch_bounds__(256) void atss_assign_kernel(
    const float* __restrict__ pred,   // [B, Q, 4] cxcywh
    const float* __restrict__ gt,     // [B, G, 4] cxcywh
    int* __restrict__ out_pred,       // [B, G*N]
    int* __restrict__ out_gt)         // [B, G*N]
{
    __shared__ float sGT[GT_TILE][4];
    __shared__ float sGnorm[GT_TILE];
    __shared__ float sD[GT_TILE][QTP];       // squared distances, one strip
    __shared__ float sTopD[GT_TILE][K_];     // sorted ascending
    __shared__ int   sTopI[GT_TILE][K_];
    __shared__ float sIoU[GT_TILE][K_];
    __shared__ int   sCnt[GT_TILE];          // per-row survivor count
    __shared__ int   sCandC[GT_TILE][CAP];   // per-row survivor columns

    const int b     = blockIdx.x / (G_ / GT_TILE);
    const int gtile = blockIdx.x % (G_ / GT_TILE);
    const int tid   = threadIdx.x;
    const int lane  = tid & 31;
    const int wave  = tid >> 5;            // 0..7
    const int hi    = lane >> 4;           // 0: lanes 0-15, 1: lanes 16-31
    const int ln    = lane & 15;

    // ---- load GT tile + norms, init top-K lists ----
    if (tid < GT_TILE * 4) {
        sGT[tid >> 2][tid & 3] = gt[(size_t)(b * G_ + gtile * GT_TILE) * 4 + tid];
    }
    __syncthreads();
    if (tid < GT_TILE) {
        float x = sGT[tid][0], y = sGT[tid][1], w = sGT[tid][2], h = sGT[tid][3];
        sGnorm[tid] = x*x + y*y + w*w + h*h;
        for (int k = 0; k < K_; ++k) { sTopD[tid][k] = 3.4e38f; sTopI[tid][k] = 0; }
    }
    __syncthreads();

    // ---- A fragment: 16x4 f32 A-matrix layout, scaled by -2 so that
    //      D = A*B + C with C = |g|^2+|p|^2 yields squared distance directly.
    v2f afrag;
    afrag.x = -2.0f * sGT[ln][hi ? 2 : 0];
    afrag.y = -2.0f * sGT[ln][hi ? 3 : 1];

    const float4* pred4 = (const float4*)(pred + (size_t)b * Q_ * 4);

    // filter-pass mapping: 16 threads per row; thread covers cols fs + 16*j
    const int fr = tid >> 4;               // row 0..15
    const int fs = tid & 15;               // sub-id 0..15

    for (int q0 = 0; q0 < Q_; q0 += QT) {
        if (tid < GT_TILE) sCnt[tid] = 0;

        // ---- Phase A: WMMA distance strip into LDS ----
        for (int st = wave; st < QT / 16; st += 8) {
            const int qbase = q0 + st * 16;
            float4 pb = pred4[qbase + ln];                   // coalesced 16B/lane
            v2f bfrag;                                       // 4x16 f32 B layout
            bfrag.x = hi ? pb.z : pb.x;
            bfrag.y = hi ? pb.w : pb.y;
            float pn = pb.x*pb.x + pb.y*pb.y + pb.z*pb.z + pb.w*pb.w;
            v8f c;
            #pragma unroll
            for (int v = 0; v < 8; ++v) c[v] = sGnorm[v + 8 * hi] + pn;
            v8f d = __builtin_amdgcn_wmma_f32_16x16x4_f32(
                        /*neg_a=*/false, afrag, /*neg_b=*/false, bfrag,
                        /*c_mod=*/(short)0, c, /*reuse_a=*/false, /*reuse_b=*/false);
            #pragma unroll
            for (int v = 0; v < 8; ++v)
                sD[v + 8 * hi][st * 16 + ln] = d[v];
        }
        __syncthreads();

        // ---- Phase B1: cooperative threshold filter (all 256 threads) ----
        // Frozen threshold per strip: stale threshold only admits extras.
        if (q0 > 0) {
            const float thr = sTopD[fr][K_ - 1];
            #pragma unroll 1
            for (int j = 0; j < QT / 16; ++j) {
                const int c = fs + 16 * j;
                if (sD[fr][c] < thr) {
                    int slot = atomicAdd(&sCnt[fr], 1);      // ds_add_rtn_u32
                    if (slot < CAP) sCandC[fr][slot] = c;
                }
            }
        }
        __syncthreads();

        // ---- Phase B2: serial merge of survivors (16 row-owner threads) ----
        if (tid < GT_TILE) {
            const int r = tid;
            float* topD = sTopD[r];
            int*   topI = sTopI[r];
            const int cnt = (q0 == 0) ? (CAP + 1) : sCnt[r];
            if (cnt > CAP) {
                // overflow / first strip: always-correct full scan
                #pragma unroll 1
                for (int c = 0; c < QT; ++c)
                    topk_insert(topD, topI, sD[r][c], q0 + c);
            } else {
                // sort survivor columns ascending (tie stability), then insert
                #pragma unroll 1
                for (int i = 1; i < cnt; ++i) {
                    int key = sCandC[r][i];
                    int j = i - 1;
                    #pragma unroll 1
                    while (j >= 0 && sCandC[r][j] > key) {
                        sCandC[r][j + 1] = sCandC[r][j];
                        --j;
                    }
                    sCandC[r][j + 1] = key;
                }
                #pragma unroll 1
                for (int i = 0; i < cnt; ++i) {
                    const int c = sCandC[r][i];
                    topk_insert(topD, topI, sD[r][c], q0 + c);
                }
            }
        }
        __syncthreads();
    }

    // ---- Phase C: IoU(gt_r, candidate_k) for all 16x32 pairs ----
    for (int idx = tid; idx < GT_TILE * K_; idx += 256) {
        const int r = idx / K_, k = idx % K_;
        float gx = sGT[r][0], gy = sGT[r][1], gw = sGT[r][2], gh = sGT[r][3];
        float gx0 = gx - 0.5f * gw, gy0 = gy - 0.5f * gh;
        float gx1 = gx + 0.5f * gw, gy1 = gy + 0.5f * gh;
        float4 pb = pred4[sTopI[r][k]];
        float px0 = pb.x - 0.5f * pb.z, py0 = pb.y - 0.5f * pb.w;
        float px1 = pb.x + 0.5f * pb.z, py1 = pb.y + 0.5f * pb.w;
        float ix0 = fmaxf(gx0, px0), iy0 = fmaxf(gy0, py0);
        float ix1 = fminf(gx1, px1), iy1 = fminf(gy1, py1);
        float iw = fmaxf(ix1 - ix0, 0.0f), ih = fmaxf(iy1 - iy0, 0.0f);
        float inter = iw * ih;
        float areaG = (gx1 - gx0) * (gy1 - gy0);
        float areaP = (px1 - px0) * (py1 - py0);
        sIoU[r][k] = inter / (areaG + areaP - inter);
    }
    __syncthreads();

    // ---- Phase D: top-9 by IoU (repeated argmax, low-k tie break), emit ----
    if (tid < GT_TILE) {
        const int r = tid;
        const int g = gtile * GT_TILE + r;
        #pragma unroll 1
        for (int n = 0; n < N_; ++n) {
            float best = -3.4e38f; int bk = 0;
            for (int k = 0; k < K_; ++k) {
                float iv = sIoU[r][k];
                if (iv > best) { best = iv; bk = k; }
            }
            sIoU[r][bk] = -3.4e38f;
            const int o = b * (G_ * N_) + g * N_ + n;
            out_pred[o] = sTopI[r][bk];
            out_gt[o]   = g;
        }
    }
}

extern "C" void kernel_launch(void* const* d_in, const int* in_sizes, int n_in,
                              void* d_out, int out_size, void* d_ws, size_t ws_size,
                              hipStream_t stream) {
    (void)in_sizes; (void)n_in; (void)d_ws; (void)ws_size; (void)out_size;
    const float* pred = (const float*)d_in[0];   // [B,Q,4] f32
    const float* gtb  = (const float*)d_in[1];   // [B,G,4] f32
    int* outp = (int*)d_out;                     // pred_idx  [B, G*N]
    int* outg = outp + B_ * G_ * N_;             // gt_idx    [B, G*N]
    dim3 grid(B_ * (G_ / GT_TILE));
    atss_assign_kernel<<<grid, 256, 0, stream>>>(pred, gtb, outp, outg);
}